// MANTIS_12515534700720
// MI455X (gfx1250) — compile-verified
//
#include <hip/hip_runtime.h>

typedef __attribute__((ext_vector_type(2))) float v2f;
typedef __attribute__((ext_vector_type(8))) float v8f;

#define Mm 48          // modes
#define Ll 48          // sites
#define Nn 256         // samples
#define MM 2304        // 48*48
#define NWAVE 9
#define NTHR 288       // 9 wave32s: one 16x16 tile of the 48x48 Gram per wave

// Each lane owns 8 elements of every 48x48 matrix using the WMMA f32 16x16
// C/D layout: element v -> row (16*ti + 8*(lane>=16) + v), col (16*tj + lane%16).

__device__ __forceinline__ void ab_site(int tid, int n, int l,
                                        const float* __restrict__ inp,
                                        const float* __restrict__ theta,
                                        const float* __restrict__ coef,
                                        float* aA, float* bA) {
  if (tid < Mm) {
    float x   = inp[n * Ll + l];
    float ang = x * (float)(tid + 1) * 1.5707963267948966f + theta[l * Mm + tid];
    float s, c;
    sincosf(ang, &s, &c);
    float cf = coef[tid];
    aA[tid] = cf * c;   // a_m = coef_m * cos(phi)
    bA[tid] = cf * s;   // b_m = coef_m * sin(phi)
  }
}

// Rank-2 Gram tile via WMMA: G = a a^T + b b^T = [a b] @ [a b]^T  (K=4, cols 2,3 zero)
__device__ __forceinline__ v8f g_tile(const float* aA, const float* bA,
                                      int ti, int tj, int lane) {
  int r = lane & 15;
  float msk = (lane < 16) ? 1.0f : 0.0f;   // K=0,1 live in lanes 0-15; K=2,3 are zero
  v2f A, B;
  A.x = msk * aA[ti * 16 + r];
  A.y = msk * bA[ti * 16 + r];
  B.x = msk * aA[tj * 16 + r];
  B.y = msk * bA[tj * 16 + r];
  v8f c = {};
  return __builtin_amdgcn_wmma_f32_16x16x4_f32(false, A, false, B, (short)0, c,
                                               false, false);
}

__global__ void __launch_bounds__(NTHR, 1)
MANTIS_zipper_kernel(const float* __restrict__ inp,    // (N,L)
                     const float* __restrict__ theta,  // (L,M)
                     const float* __restrict__ coef,   // (M,)
                     const float* __restrict__ rand_u, // (L,N)
                     float* __restrict__ meas,         // (N,L) bits as 0.0/1.0
                     float* __restrict__ pm) {         // (N,)
  extern __shared__ float smem[];
  float* ck   = smem;              // 6 suffix checkpoints: 6*MM
  float* bsuf = ck + 6 * MM;       // 8 in-block suffixes:  8*MM
  float* aA   = bsuf + 8 * MM;     // 48
  float* bA   = aA + Mm;           // 48
  float* red  = bA + Mm;           // 2*NWAVE
  float* ctrl = red + 2 * NWAVE;   // [0]=bit broadcast, [1]=denom0

  const int n     = blockIdx.x;
  const int tid   = threadIdx.x;
  const int lane  = tid & 31;
  const int wid   = tid >> 5;
  const int ti    = wid / 3;
  const int tj    = wid - 3 * ti;
  const int kcol  = tj * 16 + (lane & 15);
  const int mbase = ti * 16 + ((lane >> 4) << 3);
  const int base  = mbase * Mm + kcol;   // row-major 48x48, stride 48 per v

  // ---------------- backward pass: suffix checkpoints every 8 sites ----------
  v8f R = {1.f, 1.f, 1.f, 1.f, 1.f, 1.f, 1.f, 1.f};
#pragma unroll
  for (int v = 0; v < 8; ++v) ck[5 * MM + base + 48 * v] = 1.0f;  // S^(5) = ones
  for (int l = Ll - 1; l >= 8; --l) {
    __syncthreads();
    ab_site(tid, n, l, inp, theta, coef, aA, bA);
    __syncthreads();
    v8f G = g_tile(aA, bA, ti, tj, lane);
    R *= G;                                   // Hadamard suffix accumulate
    if ((l & 7) == 0) {
      float* dst = ck + (l / 8 - 1) * MM;     // S^(j) = prod_{l' >= 8(j+1)} G
#pragma unroll
      for (int v = 0; v < 8; ++v) dst[base + 48 * v] = R[v];
    }
  }

  // ---------------- forward zipper ------------------------------------------
  v8f P = {1.f, 1.f, 1.f, 1.f, 1.f, 1.f, 1.f, 1.f};  // Hadamard prefix (registers)
  for (int j = 0; j < 6; ++j) {
    // Expand in-block suffixes Suf_t = S^(j) o prod_{t''>t} G_{8j+t''}
    v8f Rs;
#pragma unroll
    for (int v = 0; v < 8; ++v) Rs[v] = ck[j * MM + base + 48 * v];
#pragma unroll
    for (int v = 0; v < 8; ++v) bsuf[7 * MM + base + 48 * v] = Rs[v];
    for (int t = 6; t >= 0; --t) {
      int l = j * 8 + t + 1;
      __syncthreads();
      ab_site(tid, n, l, inp, theta, coef, aA, bA);
      __syncthreads();
      v8f G = g_tile(aA, bA, ti, tj, lane);
      Rs *= G;
#pragma unroll
      for (int v = 0; v < 8; ++v) bsuf[t * MM + base + 48 * v] = Rs[v];
    }
    // Sequential measurement of the 8 sites in this block
    for (int t = 0; t < 8; ++t) {
      int l = j * 8 + t;
      __syncthreads();
      ab_site(tid, n, l, inp, theta, coef, aA, bA);
      __syncthreads();
      float ak = aA[kcol], bk = bA[kcol];
      float am[8], bm[8];
#pragma unroll
      for (int v = 0; v < 8; ++v) { am[v] = aA[mbase + v]; bm[v] = bA[mbase + v]; }
      float s0 = 0.f, s1 = 0.f;   // s0 = a^T (PoS) a, s1 = b^T (PoS) b
#pragma unroll
      for (int v = 0; v < 8; ++v) {
        float T = P[v] * bsuf[t * MM + base + 48 * v];
        s0 = fmaf(T * am[v], ak, s0);
        s1 = fmaf(T * bm[v], bk, s1);
      }
#pragma unroll
      for (int off = 16; off > 0; off >>= 1) {
        s0 += __shfl_xor(s0, off);
        s1 += __shfl_xor(s1, off);
      }
      if (lane == 0) { red[2 * wid] = s0; red[2 * wid + 1] = s1; }
      __syncthreads();
      if (tid == 0) {
        float t0 = 0.f, t1 = 0.f;
        for (int w = 0; w < NWAVE; ++w) { t0 += red[2 * w]; t1 += red[2 * w + 1]; }
        float denom = fabsf(t0 + t1);          // |inner(mps_cur)|
        if (l == 0) ctrl[1] = denom;           // |inner(mps)| for final normalization
        float p1  = fabsf(t1) / denom;
        float bit = (rand_u[l * Nn + n] < p1) ? 1.0f : 0.0f;
        ctrl[0] = bit;
        meas[n * Ll + l] = bit;
      }
      __syncthreads();
      float bit = ctrl[0];
#pragma unroll
      for (int v = 0; v < 8; ++v) {            // rank-1 Hadamard prefix update
        float f = (bit != 0.0f) ? (bm[v] * bk) : (am[v] * ak);
        P[v] *= f;
      }
    }
  }

  // ---------------- final normalized probability ----------------------------
  float sp = 0.f;
#pragma unroll
  for (int v = 0; v < 8; ++v) sp += P[v];
#pragma unroll
  for (int off = 16; off > 0; off >>= 1) sp += __shfl_xor(sp, off);
  if (lane == 0) red[wid] = sp;
  __syncthreads();
  if (tid == 0) {
    float tot = 0.f;
    for (int w = 0; w < NWAVE; ++w) tot += red[w];
    pm[n] = fabsf(tot) / ctrl[1];
  }
}

extern "C" void kernel_launch(void* const* d_in, const int* in_sizes, int n_in,
                              void* d_out, int out_size, void* d_ws, size_t ws_size,
                              hipStream_t stream) {
  (void)in_sizes; (void)n_in; (void)out_size; (void)d_ws; (void)ws_size;
  const float* inp    = (const float*)d_in[0];  // (N,L)
  const float* theta  = (const float*)d_in[1];  // (L,M)
  const float* coef   = (const float*)d_in[2];  // (M,)
  const float* rand_u = (const float*)d_in[3];  // (L,N)
  float* out  = (float*)d_out;
  float* meas = out;            // N*L measurement bits (as floats)
  float* pm   = out + Nn * Ll;  // N probabilities

  size_t shmem = (size_t)(6 * MM + 8 * MM + 2 * Mm + 2 * NWAVE + 2) * sizeof(float);
  MANTIS_zipper_kernel<<<Nn, NTHR, shmem, stream>>>(inp, theta, coef, rand_u, meas, pm);
}